// GNNModel_53618371723995
// MI455X (gfx1250) — compile-verified
//
#include <hip/hip_runtime.h>
#include <hip/hip_bf16.h>

#define N_NODES 50000
#define N_EDGES 600000
#define D_IN 128
#define D_HID 128
#define N_CLASSES 40

typedef __attribute__((ext_vector_type(2))) float v2f;
typedef __attribute__((ext_vector_type(8))) float v8f;

// ---------------------------------------------------------------------------
// degrees: out_deg via src, in_deg via dst (float accumulate, exact for counts)
// ---------------------------------------------------------------------------
__global__ void degree_kernel(const int* __restrict__ src, const int* __restrict__ dst,
                              float* __restrict__ outdeg, float* __restrict__ indeg, int E) {
    int e = blockIdx.x * blockDim.x + threadIdx.x;
    if (e >= E) return;
    atomicAdd(outdeg + src[e], 1.0f);
    atomicAdd(indeg  + dst[e], 1.0f);
}

__global__ void norm_kernel(float* __restrict__ ns, float* __restrict__ nd, int n) {
    int i = blockIdx.x * blockDim.x + threadIdx.x;
    if (i >= n) return;
    ns[i] = rsqrtf(fmaxf(ns[i], 1.0f));
    nd[i] = rsqrtf(fmaxf(nd[i], 1.0f));
}

// ---------------------------------------------------------------------------
// edge scatter, 128-dim: 32 lanes per edge, float4 per lane, L2-resident atomics
// ---------------------------------------------------------------------------
__global__ void scatter128_kernel(const float* __restrict__ H, const int* __restrict__ src,
                                  const int* __restrict__ dst, const float* __restrict__ ns,
                                  float* __restrict__ agg, int E) {
    int t = blockIdx.x * blockDim.x + threadIdx.x;
    int e = t >> 5;
    if (e >= E) return;
    int c = (t & 31) * 4;
    int s = src[e], d = dst[e];
    float w = ns[s];
    float4 v = *(const float4*)(H + (size_t)s * 128 + c);
    float* p = agg + (size_t)d * 128 + c;
    atomicAdd(p + 0, v.x * w);
    atomicAdd(p + 1, v.y * w);
    atomicAdd(p + 2, v.z * w);
    atomicAdd(p + 3, v.w * w);
}

// edge scatter, 40-dim (layer 3, weight applied first)
__global__ void scatter40_kernel(const float* __restrict__ T, const int* __restrict__ src,
                                 const int* __restrict__ dst, const float* __restrict__ ns,
                                 float* __restrict__ out, int E) {
    int e = blockIdx.x;
    int c = threadIdx.x;
    if (e >= E || c >= N_CLASSES) return;
    int s = src[e], d = dst[e];
    atomicAdd(out + (size_t)d * N_CLASSES + c, T[(size_t)s * N_CLASSES + c] * ns[s]);
}

__global__ void finalize_kernel(float* __restrict__ out, const float* __restrict__ nd,
                                const float* __restrict__ b3, int n) {
    int t = blockIdx.x * blockDim.x + threadIdx.x;
    if (t >= n) return;
    int node = t / N_CLASSES;
    int c = t - node * N_CLASSES;
    out[t] = out[t] * nd[node] + b3[c];
}

// ---------------------------------------------------------------------------
// C[nrows x 128] = relu( (A * nd_row) @ W[128x128] + b )  via v_wmma_f32_16x16x4_f32
// block = 256 threads (8 waves); wave w handles a 16-row strip.
//
// W is staged in LDS in K-pair-interleaved, bank-swizzled layout:
//   W[k][n] -> Wl[(k>>1)*256 + ((2n) ^ (((k>>1)&1)*32)) + (k&1)]
// so a B fragment (W[k0][n], W[k0+1][n]) is one contiguous 8B-aligned
// ds_load_b64 straight into an even VGPR pair (no repack movs), and the
// XOR-32 swizzle puts the laneHi=0 half-wave on banks 0-31 and the
// laneHi=1 half-wave on banks 32-63 -> conflict-free.
//
// A-frag (16x4 f32): lane L holds row = L%16, K = 4*kk + 2*(L/16) + {0,1}
// C tile (16x16 f32): vgpr i / lane L -> M = i + 8*(L/16), N = L%16
// ---------------------------------------------------------------------------
__global__ __launch_bounds__(256) void gemm128_relu_kernel(
        const float* __restrict__ A, const float* __restrict__ ndv,
        const float* __restrict__ W, const float* __restrict__ bias,
        float* __restrict__ C, int nrows) {
    __shared__ float Wl[128 * 128];   // exactly 64 KB
    for (int i = threadIdx.x; i < 128 * 128; i += 256) {
        int k = i >> 7, n = i & 127;
        int p = k >> 1;
        Wl[p * 256 + ((2 * n) ^ ((p & 1) * 32)) + (k & 1)] = W[i];
    }
    __syncthreads();

    const int wave   = threadIdx.x >> 5;
    const int lane   = threadIdx.x & 31;
    const int laneHi = lane >> 4;     // 0 or 1 (selects K pair)
    const int laneLo = lane & 15;
    const int rowBase = blockIdx.x * 128 + wave * 16;
    int row  = rowBase + laneLo;
    int rowc = row < nrows ? row : 0;          // clamp OOB reads, stores predicated
    const float s = ndv[rowc];                 // dst-norm folded into A fragment
    const float* Arow = A + (size_t)rowc * 128 + 2 * laneHi;

    const int swz = laneHi * 32;
    const float* Wbase = Wl + laneHi * 256 + 2 * laneLo;

    v8f acc[8];
    const v8f zero = {0.f, 0.f, 0.f, 0.f, 0.f, 0.f, 0.f, 0.f};
    #pragma unroll
    for (int j = 0; j < 8; ++j) acc[j] = zero;

    for (int kk = 0; kk < 32; ++kk) {
        float2 av = *(const float2*)(Arow + 4 * kk);
        v2f a; a.x = av.x * s; a.y = av.y * s;
        const float* Wk = Wbase + kk * 512;    // (2kk)*256 dwords; + laneHi row above
        #pragma unroll
        for (int j = 0; j < 8; ++j) {
            v2f b = *(const v2f*)(Wk + ((j * 32) ^ swz));
            acc[j] = __builtin_amdgcn_wmma_f32_16x16x4_f32(
                false, a, false, b, (short)0, acc[j], false, false);
        }
    }

    #pragma unroll
    for (int j = 0; j < 8; ++j) {
        const int n = j * 16 + laneLo;
        const float bv = bias[n];
        #pragma unroll
        for (int i = 0; i < 8; ++i) {
            int m = rowBase + i + laneHi * 8;
            float v = acc[j][i] + bv;
            v = v > 0.f ? v : 0.f;
            if (m < nrows) C[(size_t)m * 128 + n] = v;
        }
    }
}

// ---------------------------------------------------------------------------
// T[nrows x 40] = A[nrows x 128] @ W3[128 x 40]  (zero-padded to 48 cols)
// Packed pair-row stride is 96 dwords; 96 mod 64 banks == 32, so the two
// half-waves land on disjoint bank halves with no explicit swizzle.
//   W3[k][n] -> Wl[(k>>1)*96 + 2n + (k&1)]
// ---------------------------------------------------------------------------
__global__ __launch_bounds__(256) void gemm40_kernel(
        const float* __restrict__ A, const float* __restrict__ W3,
        float* __restrict__ T, int nrows) {
    __shared__ float Wl[64 * 96];     // 24 KB
    for (int i = threadIdx.x; i < 128 * 48; i += 256) {
        int k = i / 48, n = i - k * 48;
        float v = (n < N_CLASSES) ? W3[k * N_CLASSES + n] : 0.f;
        Wl[(k >> 1) * 96 + 2 * n + (k & 1)] = v;
    }
    __syncthreads();

    const int wave   = threadIdx.x >> 5;
    const int lane   = threadIdx.x & 31;
    const int laneHi = lane >> 4;
    const int laneLo = lane & 15;
    const int rowBase = blockIdx.x * 128 + wave * 16;
    int row  = rowBase + laneLo;
    int rowc = row < nrows ? row : 0;
    const float* Arow  = A + (size_t)rowc * 128 + 2 * laneHi;
    const float* Wbase = Wl + laneHi * 96 + 2 * laneLo;

    v8f acc[3];
    const v8f zero = {0.f, 0.f, 0.f, 0.f, 0.f, 0.f, 0.f, 0.f};
    #pragma unroll
    for (int j = 0; j < 3; ++j) acc[j] = zero;

    for (int kk = 0; kk < 32; ++kk) {
        float2 av = *(const float2*)(Arow + 4 * kk);
        v2f a; a.x = av.x; a.y = av.y;
        const float* Wk = Wbase + kk * 192;    // (2kk)*96 dwords
        #pragma unroll
        for (int j = 0; j < 3; ++j) {
            v2f b = *(const v2f*)(Wk + j * 32);
            acc[j] = __builtin_amdgcn_wmma_f32_16x16x4_f32(
                false, a, false, b, (short)0, acc[j], false, false);
        }
    }

    #pragma unroll
    for (int j = 0; j < 3; ++j) {
        const int n = j * 16 + laneLo;
        #pragma unroll
        for (int i = 0; i < 8; ++i) {
            int m = rowBase + i + laneHi * 8;
            if (m < nrows && n < N_CLASSES)
                T[(size_t)m * N_CLASSES + n] = acc[j][i];
        }
    }
}

// ---------------------------------------------------------------------------
extern "C" void kernel_launch(void* const* d_in, const int* in_sizes, int n_in,
                              void* d_out, int out_size, void* d_ws, size_t ws_size,
                              hipStream_t stream) {
    (void)in_sizes; (void)n_in; (void)out_size; (void)ws_size;
    const float* feat = (const float*)d_in[0];
    const int*   src  = (const int*)d_in[1];
    const int*   dst  = (const int*)d_in[2];
    const float* W1   = (const float*)d_in[3];
    const float* b1   = (const float*)d_in[4];
    const float* W2   = (const float*)d_in[5];
    const float* b2   = (const float*)d_in[6];
    const float* W3   = (const float*)d_in[7];
    const float* b3   = (const float*)d_in[8];
    float* out = (float*)d_out;

    // workspace layout (floats)
    float* ws   = (float*)d_ws;
    float* ns   = ws;                               // [N]
    float* nd   = ns + N_NODES;                     // [N]
    float* bufA = nd + N_NODES;                     // [N x 128] aggregation
    float* bufB = bufA + (size_t)N_NODES * 128;     // [N x 128] hidden
    float* t3   = bufB + (size_t)N_NODES * 128;     // [N x 40]  h2 @ W3

    const int nGemm = (N_NODES + 127) / 128;        // 391
    const int nScatter128 = (N_EDGES * 32 + 255) / 256;

    // degrees + norms
    hipMemsetAsync(ns, 0, sizeof(float) * 2 * N_NODES, stream);
    degree_kernel<<<(N_EDGES + 255) / 256, 256, 0, stream>>>(src, dst, ns, nd, N_EDGES);
    norm_kernel<<<(N_NODES + 255) / 256, 256, 0, stream>>>(ns, nd, N_NODES);

    // layer 1: aggregate(features) -> gemm+bias+relu
    hipMemsetAsync(bufA, 0, sizeof(float) * (size_t)N_NODES * 128, stream);
    scatter128_kernel<<<nScatter128, 256, 0, stream>>>(feat, src, dst, ns, bufA, N_EDGES);
    gemm128_relu_kernel<<<nGemm, 256, 0, stream>>>(bufA, nd, W1, b1, bufB, N_NODES);

    // layer 2: aggregate(h1) -> gemm+bias+relu
    hipMemsetAsync(bufA, 0, sizeof(float) * (size_t)N_NODES * 128, stream);
    scatter128_kernel<<<nScatter128, 256, 0, stream>>>(bufB, src, dst, ns, bufA, N_EDGES);
    gemm128_relu_kernel<<<nGemm, 256, 0, stream>>>(bufA, nd, W2, b2, bufB, N_NODES);

    // layer 3: gemm first (128 > 40), then aggregate 40-dim, then norm + bias
    gemm40_kernel<<<nGemm, 256, 0, stream>>>(bufB, W3, t3, N_NODES);
    hipMemsetAsync(out, 0, sizeof(float) * (size_t)N_NODES * N_CLASSES, stream);
    scatter40_kernel<<<N_EDGES, 64, 0, stream>>>(t3, src, dst, ns, out, N_EDGES);
    finalize_kernel<<<(N_NODES * N_CLASSES + 255) / 256, 256, 0, stream>>>(
        out, nd, b3, N_NODES * N_CLASSES);
}